// EmbeddingMPO_5952824673128
// MI455X (gfx1250) — compile-verified
//
#include <hip/hip_runtime.h>

typedef __attribute__((ext_vector_type(2))) float v2f;
typedef __attribute__((ext_vector_type(8))) float v8f;

// ---------------------------------------------------------------------------
// Workspace layout (floats):
//   P12 : [64][16][64]   = 65536   (i1i2 combo -> (o1o2) x r2)
//   S45 : [64][64][12]   = 49152   (i4i5 combo -> r3 x (o4o5))
//   P123: [512][64][64]  = 2097152 (i1i2i3 combo -> (o1o2o3) x r3)
// total ~8.85 MB
// ---------------------------------------------------------------------------

// K1: small prefix/suffix contractions (2 MMAC + 1 MMAC total) -> scalar FMA.
__global__ __launch_bounds__(256) void mpo_pre_kernel(
    const float* __restrict__ c0, const float* __restrict__ c1,
    const float* __restrict__ c3, const float* __restrict__ c4,
    float* __restrict__ P12, float* __restrict__ S45) {
  const int bid = blockIdx.x, tid = threadIdx.x;
  if (bid < 64) {
    // P12[i1i2][p=(o1,o2)][r2] = sum_r1 c0[0,i1,o1,r1] * c1[r1,i2,o2,r2]
    const int i1 = bid >> 3, i2 = bid & 7;
    for (int o = tid; o < 1024; o += 256) {
      const int p = o >> 6, r2 = o & 63;
      const int o1 = p >> 2, o2 = p & 3;
      const float* a = c0 + (i1 * 4 + o1) * 32;          // stride 1 over r1
      const float* b = c1 + (i2 * 4 + o2) * 64 + r2;     // stride 2048 over r1
      float s = 0.f;
      for (int r1 = 0; r1 < 32; ++r1) s += a[r1] * b[r1 * 2048];
      P12[bid * 1024 + o] = s;
    }
  } else {
    // S45[i4i5][r3][q=(o4,o5)] = sum_r4 c3[r3,i4,o4,r4] * c4[r4,i5,o5]
    const int cb = bid - 64;
    const int i4 = cb >> 3, i5 = cb & 7;
    for (int o = tid; o < 768; o += 256) {
      const int r3 = o / 12, q = o % 12;
      const int o4 = q / 3, o5 = q % 3;
      const float* a = c3 + ((r3 * 8 + i4) * 4 + o4) * 24;  // stride 1 over r4
      const float* b = c4 + i5 * 3 + o5;                    // stride 24 over r4
      float s = 0.f;
      for (int r4 = 0; r4 < 24; ++r4) s += a[r4] * b[r4 * 24];
      S45[cb * 768 + o] = s;
    }
  }
}

// K2: P123[combo] (16x256) = P12[c12] (16x64) @ core2[:,i3,:,:] (64x256).
// One wave per (combo, 16-wide N tile): 512*16 = 8192 waves.
__global__ __launch_bounds__(256) void mpo_p123_wmma_kernel(
    const float* __restrict__ c2, const float* __restrict__ P12,
    float* __restrict__ P123) {
  const int wave  = blockIdx.x * 8 + (threadIdx.x >> 5);
  const int lane  = threadIdx.x & 31;
  const int combo = wave >> 4;   // (i1*8+i2)*8 + i3
  const int ntile = wave & 15;
  const int c12 = combo >> 3, i3 = combo & 7;

  const float* A = P12 + c12 * 1024;     // 16x64 row-major
  const float* B = c2 + i3 * 256;        // B[k][n] = B[k*2048 + n], n=(o3,r3)

  const int m    = lane & 15;
  const int half = lane >> 4;
  const int klo  = half * 2;
  const int n    = ntile * 16 + m;

  v8f acc = {0.f, 0.f, 0.f, 0.f, 0.f, 0.f, 0.f, 0.f};
#pragma unroll
  for (int kk = 0; kk < 16; ++kk) {
    const int k0 = kk * 4 + klo;                       // even -> 8B aligned
    v2f a = *(const v2f*)(A + m * 64 + k0);
    v2f b;
    b.x = B[k0 * 2048 + n];
    b.y = B[(k0 + 1) * 2048 + n];
    acc = __builtin_amdgcn_wmma_f32_16x16x4_f32(
        /*neg_a=*/false, a, /*neg_b=*/false, b,
        /*c_mod=*/(short)0, acc, /*reuse_a=*/false, /*reuse_b=*/false);
  }

  float* D = P123 + combo * 4096;        // 16x256 row-major == [64][64]
#pragma unroll
  for (int v = 0; v < 8; ++v) {
    const int row = v + half * 8;
    D[row * 256 + n] = acc[v];
  }
}

// K3: per token, out_row(64x12) = P123[id>>6] (64x64) @ S45[id&63] (64x12),
// N padded to 16 via value-select (no divergence around the WMMAs).
// One wave per (token, 16-row M tile): 4096*4 = 16384 waves.
__global__ __launch_bounds__(256) void mpo_gather_wmma_kernel(
    const int* __restrict__ ids, const float* __restrict__ P123,
    const float* __restrict__ S45, float* __restrict__ out) {
  const int wave  = blockIdx.x * 8 + (threadIdx.x >> 5);
  const int lane  = threadIdx.x & 31;
  const int tok   = wave >> 2;
  const int mtile = wave & 3;

  const int id   = ids[tok];        // < 32128
  const int c45  = id & 63;         // i4*8 + i5
  const int c123 = id >> 6;         // (i1*8+i2)*8 + i3

  const float* A = P123 + c123 * 4096 + mtile * (16 * 64);  // 16x64 slice
  const float* B = S45 + c45 * 768;                         // 64x12 row-major

  const int m    = lane & 15;
  const int half = lane >> 4;
  const int klo  = half * 2;
  const int n    = m;                       // single N tile (cols 0..15)
  const int ncl  = (n < 12) ? n : 11;       // clamp: keep loads in-bounds
  const float msk = (n < 12) ? 1.f : 0.f;   // zero padded columns

  v8f acc = {0.f, 0.f, 0.f, 0.f, 0.f, 0.f, 0.f, 0.f};
#pragma unroll
  for (int kk = 0; kk < 16; ++kk) {
    const int k0 = kk * 4 + klo;
    v2f a = *(const v2f*)(A + m * 64 + k0);
    v2f b;
    b.x = B[k0 * 12 + ncl] * msk;
    b.y = B[(k0 + 1) * 12 + ncl] * msk;
    acc = __builtin_amdgcn_wmma_f32_16x16x4_f32(
        false, a, false, b, (short)0, acc, false, false);
  }

  if (n < 12) {
    float* o = out + tok * 768 + (mtile * 16) * 12 + n;
#pragma unroll
    for (int v = 0; v < 8; ++v) {
      const int row = v + half * 8;
      o[row * 12] = acc[v];
    }
  }
}

extern "C" void kernel_launch(void* const* d_in, const int* in_sizes, int n_in,
                              void* d_out, int out_size, void* d_ws, size_t ws_size,
                              hipStream_t stream) {
  const int*   ids = (const int*)d_in[0];     // [8,512] token ids
  const float* c0  = (const float*)d_in[1];   // (1,8,4,32)
  const float* c1  = (const float*)d_in[2];   // (32,8,4,64)
  const float* c2  = (const float*)d_in[3];   // (64,8,4,64)
  const float* c3  = (const float*)d_in[4];   // (64,8,4,24)
  const float* c4  = (const float*)d_in[5];   // (24,8,3,1)
  float* out = (float*)d_out;

  float* ws   = (float*)d_ws;
  float* P12  = ws;                           // 65536 floats
  float* S45  = ws + 65536;                   // 49152 floats
  float* P123 = ws + 65536 + 49152;           // 2097152 floats

  mpo_pre_kernel<<<128, 256, 0, stream>>>(c0, c1, c3, c4, P12, S45);
  mpo_p123_wmma_kernel<<<1024, 256, 0, stream>>>(c2, P12, P123);
  mpo_gather_wmma_kernel<<<2048, 256, 0, stream>>>(ids, P123, S45, out);
}